// PerspectiveAwareSizeConsistencyLoss_18107582120050
// MI455X (gfx1250) — compile-verified
//
#include <hip/hip_runtime.h>
#include <hip/hip_bf16.h>
#include <math.h>

// ---------------------------------------------------------------------------
// PerspectiveAwareSizeConsistencyLoss for MI455X (gfx1250, wave32, WMMA).
//
// ws float layout (offsets in floats):
//   0    cnt[16]            16   sx[16]          32   sy[16]
//   48   vx[16]             64   vy[16]
//   80   mean_x[16]         96   mean_y[16]
//   112  std_x[16]          128  std_y[16]      (80..143 contiguous stats)
//   144  den[16]            160  num[16]
//   176  M[0][16] per b     192  M[1][16] per b  208  M[16][16] per b
//   224  counts[16][15]     464  rhs[16][17]     736  theta[16][17]
//   1008 valid_fit[16]      1024 M16[16][16][16] (dense top-left block)
//   total = 5120 floats
// ---------------------------------------------------------------------------

typedef __attribute__((ext_vector_type(2))) float v2f;
typedef __attribute__((ext_vector_type(8))) float v8f;

#define WS_CNT   0
#define WS_SX    16
#define WS_SY    32
#define WS_VX    48
#define WS_VY    64
#define WS_MX    80
#define WS_MY    96
#define WS_SDX   112
#define WS_SDY   128
#define WS_DEN   144
#define WS_NUM   160
#define WS_M016  176
#define WS_M116  192
#define WS_M1616 208
#define WS_CNTS  224
#define WS_RHS   464
#define WS_THETA 736
#define WS_VALID 1008
#define WS_M16   1024
#define WS_TOTAL 5120

__global__ void pascl_zero(float* ws, int n) {
    int i = blockIdx.x * blockDim.x + threadIdx.x;
    if (i < n) ws[i] = 0.0f;
}

// Pass 1: per-batch count, sum_x, sum_y (LDS accumulate, one global flush/block)
__global__ void pascl_stats(const float* __restrict__ bbox,
                            const int* __restrict__ seg,
                            float* __restrict__ ws, int N) {
    __shared__ float sc[16], sx[16], sy[16];
    if (threadIdx.x < 16) { sc[threadIdx.x] = 0.f; sx[threadIdx.x] = 0.f; sy[threadIdx.x] = 0.f; }
    __syncthreads();
    int e = blockIdx.x * blockDim.x + threadIdx.x;
    if (e < N) {
        int b = seg[e];
        atomicAdd(&sc[b], 1.0f);
        atomicAdd(&sx[b], bbox[e * 5 + 0]);
        atomicAdd(&sy[b], bbox[e * 5 + 1]);
    }
    __syncthreads();
    int t = threadIdx.x;
    if (t < 16) {
        if (sc[t] != 0.f) atomicAdd(ws + WS_CNT + t, sc[t]);
        if (sx[t] != 0.f) atomicAdd(ws + WS_SX + t, sx[t]);
        if (sy[t] != 0.f) atomicAdd(ws + WS_SY + t, sy[t]);
    }
}

__global__ void pascl_means(float* ws) {
    int t = threadIdx.x;
    if (t < 16) {
        float n = fmaxf(ws[WS_CNT + t], 1.0f);
        ws[WS_MX + t] = ws[WS_SX + t] / n;
        ws[WS_MY + t] = ws[WS_SY + t] / n;
    }
}

// Pass 2: per-batch sum of squared deviations
__global__ void pascl_vars(const float* __restrict__ bbox,
                           const int* __restrict__ seg,
                           float* __restrict__ ws, int N) {
    __shared__ float vx[16], vy[16];
    if (threadIdx.x < 16) { vx[threadIdx.x] = 0.f; vy[threadIdx.x] = 0.f; }
    __syncthreads();
    int e = blockIdx.x * blockDim.x + threadIdx.x;
    if (e < N) {
        int b = seg[e];
        float dx = bbox[e * 5 + 0] - ws[WS_MX + b];
        float dy = bbox[e * 5 + 1] - ws[WS_MY + b];
        atomicAdd(&vx[b], dx * dx);
        atomicAdd(&vy[b], dy * dy);
    }
    __syncthreads();
    int t = threadIdx.x;
    if (t < 16) {
        if (vx[t] != 0.f) atomicAdd(ws + WS_VX + t, vx[t]);
        if (vy[t] != 0.f) atomicAdd(ws + WS_VY + t, vy[t]);
    }
}

__global__ void pascl_stds(float* ws) {
    int t = threadIdx.x;
    if (t < 16) {
        float dof = fmaxf(ws[WS_CNT + t] - 1.0f, 1.0f);
        ws[WS_SDX + t] = fmaxf(sqrtf(ws[WS_VX + t] / dof), 1e-6f);
        ws[WS_SDY + t] = fmaxf(sqrtf(ws[WS_VY + t] / dof), 1e-6f);
    }
}

struct Elem { int b, l; float p, q, sw, t, wgt; };

// stat = LDS copy of ws[80..143]: mx[16], my[16], sdx[16], sdy[16]
__device__ __forceinline__ Elem pascl_load_elem(const float* __restrict__ bbox,
                                                const float* __restrict__ scores,
                                                const int* __restrict__ labels,
                                                int b, int e,
                                                const float* stat) {
    Elem r; r.b = b;
    float x = bbox[e * 5 + 0];
    float y = bbox[e * 5 + 1];
    float w = fmaxf(bbox[e * 5 + 2], 0.01f);
    float h = fmaxf(bbox[e * 5 + 3], 0.01f);
    r.l   = labels[e];
    r.wgt = fmaxf(scores[e], 1e-6f);
    r.sw  = sqrtf(r.wgt);
    r.p   = (x - stat[b]) / stat[32 + b] * r.sw;       // A_w[0]
    r.q   = (y - stat[16 + b]) / stat[48 + b] * r.sw;  // A_w[1]
    r.t   = 0.5f * logf(w * h) * r.sw;                 // s_log * sqrt_w
    return r;
}

__device__ __forceinline__ void pascl_ex_accum(float* exb, const Elem& E) {
    atomicAdd(exb + 0,        E.p * E.t);          // rhs[0]
    atomicAdd(exb + 1,        E.q * E.t);          // rhs[1]
    atomicAdd(exb + 2 + E.l,  E.sw * E.t);         // rhs[2+l]
    atomicAdd(exb + 20,       E.wgt);              // den
    atomicAdd(exb + 21 + E.l, 1.0f);               // counts[l]
    if (E.l == 14) {                               // dim-16 interactions
        atomicAdd(exb + 17, E.p * E.sw);           // M[0][16]
        atomicAdd(exb + 18, E.q * E.sw);           // M[1][16]
        atomicAdd(exb + 19, E.sw * E.sw);          // M[16][16]
    }
}

// Pass 3: Gram/moment accumulation. Dense 16x16 block of M via
// V_WMMA_F32_16X16X4_F32 (4 elements / instruction, accumulator kept in VGPRs
// for whole same-batch runs); sparse extras via per-batch LDS atomics.
// Batch-run control flow is forced scalar (readfirstlane) so WMMA always
// executes with EXEC == all-1s via uniform s_cbranch, per ISA requirement.
__global__ void pascl_moments(const float* __restrict__ bbox,
                              const float* __restrict__ scores,
                              const int* __restrict__ labels,
                              const int* __restrict__ seg,
                              float* __restrict__ ws,
                              int N, int G, int gpw) {
    __shared__ float ex[16 * 36];   // per batch: rhs[17], m016, m116, m1616, den, counts[15]
    __shared__ float stat[64];      // mx[16], my[16], sdx[16], sdy[16]
    for (int i = threadIdx.x; i < 16 * 36; i += blockDim.x) ex[i] = 0.0f;
    if (threadIdx.x < 64) stat[threadIdx.x] = ws[WS_MX + threadIdx.x];
    __syncthreads();

    float* M16 = ws + WS_M16;
    const int lane = threadIdx.x & 31;
    const int m    = lane & 15;      // row of A / col of C this lane owns
    const int hl   = lane >> 4;      // lane-half selects K stripe
    const int wave = (blockIdx.x * blockDim.x + threadIdx.x) >> 5;

    int g0 = wave * gpw;
    int g1 = g0 + gpw; if (g1 > G) g1 = G;

    v8f c;
    #pragma unroll
    for (int r = 0; r < 8; ++r) c[r] = 0.0f;
    int cur_b = -1;   // scalar (derived only from readfirstlane values)

    for (int g = g0; g < g1; ++g) {
        const int base = g * 4;
        const int4 bq = *(const int4*)(seg + base);   // 16B aligned, wave-uniform
        const int b0 = __builtin_amdgcn_readfirstlane(bq.x);
        const int b1 = __builtin_amdgcn_readfirstlane(bq.y);
        const int b2 = __builtin_amdgcn_readfirstlane(bq.z);
        const int b3 = __builtin_amdgcn_readfirstlane(bq.w);

        // this lane supplies a[m] for elements eA (VGPR0 K-slot) and eB (VGPR1)
        const int eA = base + 2 * hl;
        const int eB = eA + 1;
        const int bA = hl ? b2 : b0;
        const int bB = hl ? b3 : b1;
        Elem EA = pascl_load_elem(bbox, scores, labels, bA, eA, stat);
        Elem EB = pascl_load_elem(bbox, scores, labels, bB, eB, stat);

        float aA = (m == 0) ? EA.p : (m == 1) ? EA.q : ((m - 2) == EA.l ? EA.sw : 0.0f);
        float aB = (m == 0) ? EB.p : (m == 1) ? EB.q : ((m - 2) == EB.l ? EB.sw : 0.0f);

        // sparse extras: lane 0 owns e0,e1; lane 16 owns e2,e3
        if (m == 0) {
            pascl_ex_accum(ex + EA.b * 36, EA);
            pascl_ex_accum(ex + EB.b * 36, EB);
        }

        // iterate distinct (sorted) batches inside the 4-group; mask + WMMA
        int gb = b0;
        for (;;) {
            if (gb != cur_b) {                 // scalar compare -> s_cbranch
                if (cur_b >= 0) {
                    float* Mb = M16 + cur_b * 256;
                    #pragma unroll
                    for (int r = 0; r < 8; ++r)
                        atomicAdd(Mb + (r + 8 * hl) * 16 + m, c[r]);
                }
                #pragma unroll
                for (int r = 0; r < 8; ++r) c[r] = 0.0f;
                cur_b = gb;
            }
            v2f a;
            a.x = (bA == gb) ? aA : 0.0f;
            a.y = (bB == gb) ? aB : 0.0f;
            // D(16x16,f32) += A(16x4,f32) x B(4x16,f32); symmetric => A==B regs
            c = __builtin_amdgcn_wmma_f32_16x16x4_f32(
                    false, a, false, a, (short)0, c, false, false);
            if (b3 == gb) break;               // scalar -> uniform branch
            gb = (b1 > gb) ? b1 : ((b2 > gb) ? b2 : b3);
        }
    }
    if (cur_b >= 0) {
        float* Mb = M16 + cur_b * 256;
        #pragma unroll
        for (int r = 0; r < 8; ++r)
            atomicAdd(Mb + (r + 8 * hl) * 16 + m, c[r]);
    }

    __syncthreads();
    // flush sparse extras to global
    for (int i = threadIdx.x; i < 16 * 36; i += blockDim.x) {
        float v = ex[i];
        if (v != 0.0f) {
            int b = i / 36, o = i % 36;
            float* dst;
            if      (o < 17)  dst = ws + WS_RHS   + b * 17 + o;
            else if (o == 17) dst = ws + WS_M016  + b;
            else if (o == 18) dst = ws + WS_M116  + b;
            else if (o == 19) dst = ws + WS_M1616 + b;
            else if (o == 20) dst = ws + WS_DEN   + b;
            else              dst = ws + WS_CNTS  + b * 15 + (o - 21);
            atomicAdd(dst, v);
        }
    }

    // scalar tail for N % 4 leftovers (rare; direct global atomics)
    int gid = blockIdx.x * blockDim.x + threadIdx.x;
    int ntail = N - G * 4;
    if (gid < ntail) {
        int e = G * 4 + gid;
        int b = seg[e];
        Elem E = pascl_load_elem(bbox, scores, labels, b, e, stat);
        float* Mb = M16 + b * 256;
        atomicAdd(Mb + 0,  E.p * E.p);
        atomicAdd(Mb + 1,  E.p * E.q);
        atomicAdd(Mb + 16, E.q * E.p);
        atomicAdd(Mb + 17, E.q * E.q);
        int j = 2 + E.l;
        if (j < 16) {
            atomicAdd(Mb + 0 * 16 + j, E.p * E.sw);
            atomicAdd(Mb + j * 16 + 0, E.p * E.sw);
            atomicAdd(Mb + 1 * 16 + j, E.q * E.sw);
            atomicAdd(Mb + j * 16 + 1, E.q * E.sw);
            atomicAdd(Mb + j * 16 + j, E.sw * E.sw);
        } else {
            atomicAdd(ws + WS_M016 + b,  E.p * E.sw);
            atomicAdd(ws + WS_M116 + b,  E.q * E.sw);
            atomicAdd(ws + WS_M1616 + b, E.sw * E.sw);
        }
        atomicAdd(ws + WS_RHS + b * 17 + 0,       E.p * E.t);
        atomicAdd(ws + WS_RHS + b * 17 + 1,       E.q * E.t);
        atomicAdd(ws + WS_RHS + b * 17 + 2 + E.l, E.sw * E.t);
        atomicAdd(ws + WS_DEN + b, E.wgt);
        atomicAdd(ws + WS_CNTS + b * 15 + E.l, 1.0f);
    }
}

// 16 batched 17x17 solves: Gauss-Jordan with partial pivoting, one wave/batch.
__global__ void pascl_solve(float* __restrict__ ws) {
    __shared__ float Aug[17][18];
    __shared__ int piv;
    const int b = blockIdx.x;
    const int t = threadIdx.x;
    const float* M16 = ws + WS_M16 + b * 256;

    for (int idx = t; idx < 17 * 18; idx += 32) {
        int i = idx / 18, j = idx % 18;
        float v;
        if (j == 17)                 v = ws[WS_RHS + b * 17 + i];
        else if (i < 16 && j < 16)   v = M16[i * 16 + j];
        else if (i == 16 && j == 16) v = ws[WS_M1616 + b];
        else if (i == 16)            v = (j == 0) ? ws[WS_M016 + b] : (j == 1) ? ws[WS_M116 + b] : 0.0f;
        else                         v = (i == 0) ? ws[WS_M016 + b] : (i == 1) ? ws[WS_M116 + b] : 0.0f;
        if (i == j) v += 1e-4f;      // RIDGE
        Aug[i][j] = v;
    }
    __syncthreads();

    for (int k = 0; k < 17; ++k) {
        if (t == 0) {
            int p = k; float best = fabsf(Aug[k][k]);
            for (int r = k + 1; r < 17; ++r) {
                float v = fabsf(Aug[r][k]);
                if (v > best) { best = v; p = r; }
            }
            piv = p;
        }
        __syncthreads();
        int p = piv;
        if (p != k && t < 18) {
            float tmp = Aug[k][t]; Aug[k][t] = Aug[p][t]; Aug[p][t] = tmp;
        }
        __syncthreads();
        if (t < 17 && t != k) {
            float f = Aug[t][k] / Aug[k][k];
            for (int j = k; j < 18; ++j) Aug[t][j] -= f * Aug[k][j];
        }
        __syncthreads();
    }
    if (t < 17) ws[WS_THETA + b * 17 + t] = Aug[t][17] / Aug[t][t];
    if (t == 0) {
        int K = 0;
        for (int l = 0; l < 15; ++l) K += (ws[WS_CNTS + b * 15 + l] > 0.0f) ? 1 : 0;
        ws[WS_VALID + b] = (ws[WS_CNT + b] >= (float)K + 3.0f) ? 1.0f : 0.0f;
    }
}

// Pass 4: smooth-L1 residual, weighted per-batch numerator
__global__ void pascl_loss(const float* __restrict__ bbox,
                           const float* __restrict__ scores,
                           const int* __restrict__ labels,
                           const int* __restrict__ seg,
                           float* __restrict__ ws, int N) {
    __shared__ float snum[16];
    if (threadIdx.x < 16) snum[threadIdx.x] = 0.0f;
    __syncthreads();
    int e = blockIdx.x * blockDim.x + threadIdx.x;
    if (e < N) {
        int b = seg[e];
        float x = bbox[e * 5 + 0];
        float y = bbox[e * 5 + 1];
        float w = fmaxf(bbox[e * 5 + 2], 0.01f);
        float h = fmaxf(bbox[e * 5 + 3], 0.01f);
        float slog = 0.5f * logf(w * h);
        float wgt  = fmaxf(scores[e], 1e-6f);
        int   l    = labels[e];
        float xn = (x - ws[WS_MX + b]) / ws[WS_SDX + b];
        float yn = (y - ws[WS_MY + b]) / ws[WS_SDY + b];
        const float* th = ws + WS_THETA + b * 17;
        float yhat = th[0] * xn + th[1] * yn + th[2 + l];
        float d = slog - yhat, ad = fabsf(d);
        float le = (ad < 1.0f) ? 0.5f * d * d : ad - 0.5f;   // BETA = 1
        atomicAdd(&snum[b], le * wgt);
    }
    __syncthreads();
    if (threadIdx.x < 16) {
        float v = snum[threadIdx.x];
        if (v != 0.0f) atomicAdd(ws + WS_NUM + threadIdx.x, v);
    }
}

__global__ void pascl_final(const float* __restrict__ ws, float* __restrict__ out) {
    __shared__ float tot[16], vld[16];
    int t = threadIdx.x;
    if (t < 16) {
        float li = ws[WS_NUM + t] / (ws[WS_DEN + t] + 1e-6f);
        bool valid = (ws[WS_VALID + t] > 0.5f) && (li > 0.0f);
        tot[t] = valid ? li : 0.0f;
        vld[t] = valid ? 1.0f : 0.0f;
    }
    __syncthreads();
    if (t == 0) {
        float s = 0.0f, n = 0.0f;
        for (int i = 0; i < 16; ++i) { s += tot[i]; n += vld[i]; }
        out[0] = (n > 0.0f) ? s / n : 0.0f;   // LOSS_WEIGHT = 1
    }
}

extern "C" void kernel_launch(void* const* d_in, const int* in_sizes, int n_in,
                              void* d_out, int out_size, void* d_ws, size_t ws_size,
                              hipStream_t stream) {
    const float* bbox   = (const float*)d_in[0];
    const float* scores = (const float*)d_in[1];
    const int*   labels = (const int*)d_in[2];
    const int*   seg    = (const int*)d_in[3];
    float* ws  = (float*)d_ws;
    float* out = (float*)d_out;

    const int N  = in_sizes[1];
    const int nb = (N + 255) / 256;

    pascl_zero<<<(WS_TOTAL + 255) / 256, 256, 0, stream>>>(ws, WS_TOTAL);
    pascl_stats<<<nb, 256, 0, stream>>>(bbox, seg, ws, N);
    pascl_means<<<1, 32, 0, stream>>>(ws);
    pascl_vars<<<nb, 256, 0, stream>>>(bbox, seg, ws, N);
    pascl_stds<<<1, 32, 0, stream>>>(ws);

    const int G = N / 4;                 // full 4-element WMMA groups
    const int blocks5 = 128;             // 1024 wave32 waves
    const int waves = blocks5 * (256 / 32);
    const int gpw = (G + waves - 1) / waves;
    pascl_moments<<<blocks5, 256, 0, stream>>>(bbox, scores, labels, seg, ws, N, G, gpw > 0 ? gpw : 1);

    pascl_solve<<<16, 32, 0, stream>>>(ws);
    pascl_loss<<<nb, 256, 0, stream>>>(bbox, scores, labels, seg, ws, N);
    pascl_final<<<1, 32, 0, stream>>>(ws, out);
}